// GaussianDistribution_1434519077353
// MI455X (gfx1250) — compile-verified
//
#include <hip/hip_runtime.h>
#include <cstdint>

typedef float f4 __attribute__((ext_vector_type(4)));

constexpr unsigned Bb = 64;
constexpr unsigned Nn = 16384;   // power of two: b = g>>14, n = g & 16383
constexpr unsigned Ss = 8;

// Padded per-gaussian LDS slot: 24 floats of eps + 4 floats pad = 112 bytes.
// 28-float stride => conflict-free wave32 ds_load_b128 (banks 28*t mod 64 are
// all-distinct over t=0..15 and t=16..31).
constexpr unsigned SLOT_F = 28;   // floats
constexpr unsigned SLOT_B = 112;  // bytes

__global__ __launch_bounds__(256) void gauss_sample_kernel(
    const float* __restrict__ mean,
    const float* __restrict__ scale,
    const float* __restrict__ rot,
    const float* __restrict__ eps,
    float* __restrict__ out)
{
    // 256 gaussians/block * 112 B = 28 KB LDS staging buffer
    __shared__ __align__(16) float smem[256 * SLOT_F];

    const unsigned tid   = threadIdx.x;
    const unsigned lane  = tid & 31u;
    const unsigned wv    = tid >> 5u;
    const unsigned gBase = blockIdx.x * 256u;      // first gaussian of block
    const unsigned g     = gBase + tid;            // p = b*N + n
    const unsigned b     = g >> 14;
    const unsigned n     = g & (Nn - 1u);

    // ---------------- async-stage epsilon into LDS (CDNA5 ASYNCcnt path) ----
    // Wave wv owns gaussians [gBase+wv*32, gBase+wv*32+32): 3 KB of eps.
    // Global side: perfectly coalesced b128 chunks f = 32k + lane.
    // LDS side: chunk f -> gaussian f/6, chunk f%6, in the padded slot.
    // Strength-reduced: 32k = 6*q_k + r_k, q={0,5,10,16,21,26}, r={0,2,4,...}
    //   lAddr_k = (ldsBase + wvLds + 16*lane) + 16*d_{r_k} + (112*q_k + 16*r_k)
    //   vOff_k  = (epsByteBase + wv*3072 + 16*lane) + 512*k
    // with d_r = floor((lane + r)/6): only 3 tiny divisions for all 6 copies.
    const unsigned ldsBase = (unsigned)(uintptr_t)&smem[0];   // low 32 bits = LDS offset
    {
        const unsigned d0 = lane / 6u;
        const unsigned d2 = (lane + 2u) / 6u;
        const unsigned d4 = (lane + 4u) / 6u;
        const unsigned tL = ldsBase + wv * (32u * SLOT_B) + lane * 16u;
        const unsigned tG = gBase * 96u + wv * 3072u + lane * 16u;
        const unsigned a0 = d0 * 16u, a2 = d2 * 16u, a4 = d4 * 16u;

        const unsigned lA[6] = {
            tL + a0 + ( 0u * SLOT_B +  0u),
            tL + a2 + ( 5u * SLOT_B + 32u),
            tL + a4 + (10u * SLOT_B + 64u),
            tL + a0 + (16u * SLOT_B +  0u),
            tL + a2 + (21u * SLOT_B + 32u),
            tL + a4 + (26u * SLOT_B + 64u),
        };
#pragma unroll
        for (unsigned k = 0; k < 6; ++k) {
            const unsigned vOff = tG + k * 512u;   // byte offset (GVS: saddr + vaddr)
            asm volatile("global_load_async_to_lds_b128 %0, %1, %2 th:TH_LOAD_NT"
                         :: "v"(lA[k]), "v"(vOff), "s"(eps) : "memory");
        }
    }

    // ---------------- per-gaussian parameters (coalesced NT b32 loads) ------
    const unsigned msBase = b * (3u * Nn) + n;
    const float m0 = __builtin_nontemporal_load(mean + msBase);
    const float m1 = __builtin_nontemporal_load(mean + msBase + Nn);
    const float m2 = __builtin_nontemporal_load(mean + msBase + 2u * Nn);
    const float s0 = __builtin_nontemporal_load(scale + msBase);
    const float s1 = __builtin_nontemporal_load(scale + msBase + Nn);
    const float s2 = __builtin_nontemporal_load(scale + msBase + 2u * Nn);
    const unsigned rBase = b * (4u * Nn) + n;
    float qr = __builtin_nontemporal_load(rot + rBase);
    float qx = __builtin_nontemporal_load(rot + rBase + Nn);
    float qy = __builtin_nontemporal_load(rot + rBase + 2u * Nn);
    float qz = __builtin_nontemporal_load(rot + rBase + 3u * Nn);

    // normalize quaternion
    const float inv = rsqrtf(qr*qr + qx*qx + qy*qy + qz*qz);
    qr *= inv; qx *= inv; qy *= inv; qz *= inv;

    // rotation matrix
    const float R00 = 1.f - 2.f*(qy*qy + qz*qz);
    const float R01 = 2.f*(qx*qy - qr*qz);
    const float R02 = 2.f*(qx*qz + qr*qy);
    const float R10 = 2.f*(qx*qy + qr*qz);
    const float R11 = 1.f - 2.f*(qx*qx + qz*qz);
    const float R12 = 2.f*(qy*qz - qr*qx);
    const float R20 = 2.f*(qx*qz - qr*qy);
    const float R21 = 2.f*(qy*qz + qr*qx);
    const float R22 = 1.f - 2.f*(qx*qx + qy*qy);

    // L = R * diag(s)
    const float L00 = R00*s0, L01 = R01*s1, L02 = R02*s2;
    const float L10 = R10*s0, L11 = R11*s1, L12 = R12*s2;
    const float L20 = R20*s0, L21 = R21*s1, L22 = R22*s2;

    // cov = L * L^T (symmetric: 6 unique terms)
    const float c00 = L00*L00 + L01*L01 + L02*L02;
    const float c01 = L00*L10 + L01*L11 + L02*L12;
    const float c02 = L00*L20 + L01*L21 + L02*L22;
    const float c11 = L10*L10 + L11*L11 + L12*L12;
    const float c12 = L10*L20 + L11*L21 + L12*L22;
    const float c22 = L20*L20 + L21*L21 + L22*L22;

    // wait for this wave's async copies (data is wave-private -> no barrier)
    asm volatile("s_wait_asynccnt 0" ::: "memory");

    // read my 24 eps values back: 6 x conflict-free ds_load_b128
    float e[24];
    {
        const f4* sm4 = (const f4*)&smem[tid * SLOT_F];
        f4* ev = (f4*)e;
#pragma unroll
        for (int k = 0; k < 6; ++k) ev[k] = sm4[k];
    }

    // variance[s][j] = sum_i eps[s][i] * cov[i][j]  (cov symmetric)
    float o0[8], o1[8], o2[8];
#pragma unroll
    for (int s = 0; s < 8; ++s) {
        const float ex = e[3*s + 0];
        const float ey = e[3*s + 1];
        const float ez = e[3*s + 2];
        float v0 = c00*ex + c01*ey + c02*ez;
        float v1 = c01*ex + c11*ey + c12*ez;
        float v2 = c02*ex + c12*ey + c22*ez;
        v0 = fminf(fmaxf(v0, -1.f), 1.f) + m0;
        v1 = fminf(fmaxf(v1, -1.f), 1.f) + m1;
        v2 = fminf(fmaxf(v2, -1.f), 1.f) + m2;
        o0[s] = v0; o1[s] = v1; o2[s] = v2;
    }

    // out[b, j, n*S + s]  — streamed, non-temporal b128 stores
    const size_t NS = (size_t)Nn * Ss;
    float* obase = out + (size_t)b * 3u * NS + (size_t)n * Ss;
    {
        f4 lo, hi; f4* p;
        p = (f4*)(obase);
        lo = (f4){o0[0], o0[1], o0[2], o0[3]};
        hi = (f4){o0[4], o0[5], o0[6], o0[7]};
        __builtin_nontemporal_store(lo, p);
        __builtin_nontemporal_store(hi, p + 1);
        p = (f4*)(obase + NS);
        lo = (f4){o1[0], o1[1], o1[2], o1[3]};
        hi = (f4){o1[4], o1[5], o1[6], o1[7]};
        __builtin_nontemporal_store(lo, p);
        __builtin_nontemporal_store(hi, p + 1);
        p = (f4*)(obase + 2u * NS);
        lo = (f4){o2[0], o2[1], o2[2], o2[3]};
        hi = (f4){o2[4], o2[5], o2[6], o2[7]};
        __builtin_nontemporal_store(lo, p);
        __builtin_nontemporal_store(hi, p + 1);
    }
}

extern "C" void kernel_launch(void* const* d_in, const int* in_sizes, int n_in,
                              void* d_out, int out_size, void* d_ws, size_t ws_size,
                              hipStream_t stream) {
    const float* mean  = (const float*)d_in[0];
    const float* scale = (const float*)d_in[1];
    const float* rot   = (const float*)d_in[2];
    const float* eps   = (const float*)d_in[3];
    // d_in[4] = num_samples (int, ==8) — baked into the kernel as Ss.
    float* out = (float*)d_out;

    const unsigned total  = Bb * Nn;          // 1,048,576 gaussians
    const unsigned blocks = total / 256u;     // 4096, exact
    gauss_sample_kernel<<<dim3(blocks), dim3(256), 0, stream>>>(mean, scale, rot, eps, out);
}